// HebbianNetwork_22153441313285
// MI455X (gfx1250) — compile-verified
//
#include <hip/hip_runtime.h>
#include <math.h>

// Problem constants (match reference)
#define B_     64
#define ISIZE_ 512
#define R_     1024
#define NACT_  256
#define CLIP_  2.0f
#define RCH_   4              // r-dimension split for pass 1
#define RCHLEN_ (R_ / RCH_)   // 256 rows per chunk

typedef __attribute__((ext_vector_type(2))) float v2f;
typedef __attribute__((ext_vector_type(4))) float f4;
typedef __attribute__((ext_vector_type(8))) float v8f;

// ---------------------------------------------------------------------------
// Pass 1a: partial[c,b,s] = sum_{r in chunk c} rv[b,r]*(w[r,s]+alpha[r,s]*hebb[b,r,s])
// grid = (RCH_, B), block = 256 threads, each thread owns 4 consecutive s
// (float4/B128 traffic). hebb streamed with NT loads so it never thrashes
// the 192 MB L2; w/alpha (8 MB) stay L2-resident across all 64 batches.
// ---------------------------------------------------------------------------
__global__ __launch_bounds__(256)
void rec_in_partial_kernel(const float* __restrict__ rv,
                           const float* __restrict__ w,
                           const float* __restrict__ alpha,
                           const float* __restrict__ hebb,
                           float* __restrict__ partial)
{
    const int chunk = blockIdx.x;      // 0..RCH_-1
    const int b     = blockIdx.y;
    const int t     = threadIdx.x;     // float4 index over s (s = 4*t)

    __shared__ float rvs[RCHLEN_];
    rvs[t] = rv[b * R_ + chunk * RCHLEN_ + t];   // blockDim == RCHLEN_
    __syncthreads();

    const size_t rowbase = (size_t)chunk * RCHLEN_ * (R_ / 4);
    const f4* hb = (const f4*)hebb + (size_t)b * R_ * (R_ / 4) + rowbase;
    const f4* wp = (const f4*)w     + rowbase;
    const f4* ap = (const f4*)alpha + rowbase;

    f4 acc = {0.f, 0.f, 0.f, 0.f};
#pragma unroll 4
    for (int r = 0; r < RCHLEN_; ++r) {
        const size_t off = (size_t)r * (R_ / 4) + t;
        const f4 h  = __builtin_nontemporal_load(hb + off);
        const f4 av = ap[off];
        const f4 wv = wp[off];
        const float rr = rvs[r];
        acc.x = fmaf(rr, fmaf(av.x, h.x, wv.x), acc.x);
        acc.y = fmaf(rr, fmaf(av.y, h.y, wv.y), acc.y);
        acc.z = fmaf(rr, fmaf(av.z, h.z, wv.z), acc.z);
        acc.w = fmaf(rr, fmaf(av.w, h.w, wv.w), acc.w);
    }
    f4* po = (f4*)partial + ((size_t)chunk * B_ + b) * (R_ / 4);
    po[t] = acc;
}

// ---------------------------------------------------------------------------
// Pass 1b: rec_in = sum over the RCH_ partials (1 MB -> 256 KB, trivial).
// grid = B*R/4/256 = 64 blocks. Deterministic (no float atomics).
// ---------------------------------------------------------------------------
__global__ __launch_bounds__(256)
void rec_in_reduce_kernel(const float* __restrict__ partial,
                          float* __restrict__ rec_in)
{
    const int i = blockIdx.x * 256 + threadIdx.x;   // float4 index, B*R/4 total
    const f4* p = (const f4*)partial;
    const int stride = B_ * R_ / 4;
    const f4 a = p[i];
    const f4 b = p[i + stride];
    const f4 c = p[i + 2 * stride];
    const f4 d = p[i + 3 * stride];
    f4 o;
    o.x = (a.x + b.x) + (c.x + d.x);
    o.y = (a.y + b.y) + (c.y + d.y);
    o.z = (a.z + b.z) + (c.z + d.z);
    o.w = (a.w + b.w) + (c.w + d.w);
    ((f4*)rec_in)[i] = o;
}

// ---------------------------------------------------------------------------
// WMMA fp32 GEMM: out[m,n] = act( sum_k A[m,k]*W[n,k] + bias[n] (+ extra[m,n]) )
// A: [64 x K] row-major, W: [N x K] row-major (torch Linear convention).
// One wave32 per 16x16 output tile; K-loop in steps of 4 with
// v_wmma_f32_16x16x4_f32 (exact fp32 math).
//
// A-fragment (32-bit A 16x4, ISA 7.12.2): lane&15 = M; lanes 0-15 carry
// K = kbase+{0,1}, lanes 16-31 carry K = kbase+{2,3} -> v2f at arow+kbase+2*half.
// B-fragment mirrors it with N = lane&15 on rows of W.
// C/D: VGPR i -> row = 8*half + i, col = lane&15.
// ---------------------------------------------------------------------------
__global__ __launch_bounds__(256)
void gemm_wmma_f32(const float* __restrict__ A,
                   const float* __restrict__ W,
                   const float* __restrict__ bias,
                   const float* __restrict__ extra,
                   float* __restrict__ out,
                   int K, int N, int do_tanh)
{
    const int wave = (blockIdx.x * 256 + (int)threadIdx.x) >> 5;
    const int lane = threadIdx.x & 31;
    const int ntn  = N >> 4;
    const int tm   = wave / ntn;
    const int tn   = wave - tm * ntn;

    const int half = lane >> 4;
    const int l15  = lane & 15;

    const float* arow = A + (size_t)(tm * 16 + l15) * K + half * 2;
    const float* wrow = W + (size_t)(tn * 16 + l15) * K + half * 2;

    v8f c = {0.f, 0.f, 0.f, 0.f, 0.f, 0.f, 0.f, 0.f};
    for (int k = 0; k < K; k += 4) {
        v2f a = *(const v2f*)(arow + k);
        v2f b = *(const v2f*)(wrow + k);
        // 8 args: (neg_a, A, neg_b, B, c_mod, C, reuse_a, reuse_b)
        c = __builtin_amdgcn_wmma_f32_16x16x4_f32(false, a, false, b,
                                                  (short)0, c, false, false);
    }

    const int   col  = tn * 16 + l15;
    const float bcol = bias[col];
#pragma unroll
    for (int i = 0; i < 8; ++i) {
        const int row = tm * 16 + half * 8 + i;
        float v = c[i] + bcol;
        if (extra)   v += extra[(size_t)row * N + col];
        if (do_tanh) v = tanhf(v);
        out[(size_t)row * N + col] = v;
    }
}

// ---------------------------------------------------------------------------
// g[b,s] = (tanh(dot(rlv[b,:], h2mod_W) + h2mod_b) * mf_W[s] + mf_b[s]) * rlv[b,s]
// One block per batch; LDS tree reduction for the dot product.
// ---------------------------------------------------------------------------
__global__ __launch_bounds__(256)
void eta_g_kernel(const float* __restrict__ rlv,
                  const float* __restrict__ h2mod_W,
                  const float* __restrict__ h2mod_b,
                  const float* __restrict__ mf_W,
                  const float* __restrict__ mf_b,
                  float* __restrict__ g)
{
    const int b   = blockIdx.x;
    const int tid = threadIdx.x;
    __shared__ float red[256];

    float acc = 0.f;
    for (int s = tid; s < R_; s += 256)
        acc = fmaf(rlv[b * R_ + s], h2mod_W[s], acc);
    red[tid] = acc;
    __syncthreads();
    for (int off = 128; off > 0; off >>= 1) {
        if (tid < off) red[tid] += red[tid + off];
        __syncthreads();
    }
    const float myeta = tanhf(red[0] + h2mod_b[0]);

    for (int s = tid; s < R_; s += 256) {
        const float eta = fmaf(myeta, mf_W[s], mf_b[s]);
        g[b * R_ + s] = eta * rlv[b * R_ + s];
    }
}

// ---------------------------------------------------------------------------
// Pass 2: hebb_out[b,r,s] = clip(hebb[b,r,s] + rv[b,r]*g[b,s], -2, 2)
// One block per (b,r) row; 256 lanes x float4 = full 1024-wide row.
// Pure NT stream: 268 MB in + 268 MB out of B128 traffic.
// ---------------------------------------------------------------------------
__global__ __launch_bounds__(256)
void hebb_update_kernel(const float* __restrict__ hebb,
                        const float* __restrict__ rv,
                        const float* __restrict__ g,
                        float* __restrict__ hebb_out)
{
    const int br = blockIdx.x;          // br = b*R + r
    const int b  = br >> 10;
    const float f = rv[br];

    const f4* hin  = (const f4*)hebb + (size_t)br * (R_ / 4);
    const f4* gp   = (const f4*)g    + (size_t)b  * (R_ / 4);
    f4*       hout = (f4*)hebb_out   + (size_t)br * (R_ / 4);

    const int t = threadIdx.x;
    const f4 hv = __builtin_nontemporal_load(hin + t);
    const f4 gg = gp[t];

    f4 o;
    o.x = fminf(CLIP_, fmaxf(-CLIP_, fmaf(f, gg.x, hv.x)));
    o.y = fminf(CLIP_, fmaxf(-CLIP_, fmaf(f, gg.y, hv.y)));
    o.z = fminf(CLIP_, fmaxf(-CLIP_, fmaf(f, gg.z, hv.z)));
    o.w = fminf(CLIP_, fmaxf(-CLIP_, fmaf(f, gg.w, hv.w)));
    __builtin_nontemporal_store(o, hout + t);
}

// ---------------------------------------------------------------------------
extern "C" void kernel_launch(void* const* d_in, const int* in_sizes, int n_in,
                              void* d_out, int out_size, void* d_ws, size_t ws_size,
                              hipStream_t stream)
{
    (void)in_sizes; (void)n_in; (void)out_size; (void)ws_size;

    const float* inputs = (const float*)d_in[0];
    const float* rv     = (const float*)d_in[1];
    const float* hebb   = (const float*)d_in[2];
    const float* W0 = (const float*)d_in[3];  const float* b0 = (const float*)d_in[4];
    const float* W1 = (const float*)d_in[5];  const float* b1 = (const float*)d_in[6];
    const float* W2 = (const float*)d_in[7];  const float* b2 = (const float*)d_in[8];
    const float* W3 = (const float*)d_in[9];  const float* b3 = (const float*)d_in[10];
    const float* wpl   = (const float*)d_in[11];
    const float* alpha = (const float*)d_in[12];
    const float* recW  = (const float*)d_in[13]; const float* recb = (const float*)d_in[14];
    const float* h2mW  = (const float*)d_in[15]; const float* h2mb = (const float*)d_in[16];
    const float* mfW   = (const float*)d_in[17]; const float* mfb  = (const float*)d_in[18];

    // d_out: act_dis [64,256] ++ rec_layer_values [64,1024] ++ hebb_return [64,1024,1024]
    float* out_act  = (float*)d_out;
    float* out_rlv  = out_act + B_ * NACT_;
    float* out_hebb = out_rlv + B_ * R_;

    // workspace: partial [4,64,1024], rec_in, h0, h1, h2, g  (~2.25 MB)
    float* ws      = (float*)d_ws;
    float* partial = ws;                         // RCH_*B_*R_ = 262144 floats
    float* rec_in  = partial + RCH_ * B_ * R_;
    float* h0      = rec_in + B_ * R_;
    float* h1      = h0 + B_ * R_;
    float* h2      = h1 + B_ * R_;
    float* g       = h2 + B_ * R_;

    // Pass 1 over hebb (B128 NT stream, 256 blocks)
    rec_in_partial_kernel<<<dim3(RCH_, B_), 256, 0, stream>>>(rv, wpl, alpha, hebb, partial);
    rec_in_reduce_kernel<<<(B_ * R_ / 4) / 256, 256, 0, stream>>>(partial, rec_in);

    // Dense chain (WMMA fp32). tiles = (64/16)*(N/16) waves; 8 waves/block.
    const int blk1024 = (4 * (R_ / 16)) * 32 / 256;    // 32 blocks for N=1024
    const int blk256  = (4 * (NACT_ / 16)) * 32 / 256; // 8 blocks for N=256
    gemm_wmma_f32<<<blk1024, 256, 0, stream>>>(inputs, W0, b0, rec_in, h0, ISIZE_, R_, 1);
    gemm_wmma_f32<<<blk1024, 256, 0, stream>>>(h0, W1, b1, nullptr, h1, R_, R_, 1);
    gemm_wmma_f32<<<blk1024, 256, 0, stream>>>(h1, recW, recb, nullptr, out_rlv, R_, R_, 0);
    gemm_wmma_f32<<<blk1024, 256, 0, stream>>>(h1, W2, b2, nullptr, h2, R_, R_, 1);
    gemm_wmma_f32<<<blk256,  256, 0, stream>>>(h2, W3, b3, nullptr, out_act, R_, NACT_, 0);

    // Neuromodulation gain
    eta_g_kernel<<<B_, 256, 0, stream>>>(out_rlv, h2mW, h2mb, mfW, mfb, g);

    // Pass 2 over hebb (stream in + out)
    hebb_update_kernel<<<B_ * R_, 256, 0, stream>>>(hebb, rv, g, out_hebb);
}